// MergedEmbeddingBagModel_61186104099185
// MI455X (gfx1250) — compile-verified
//
#include <hip/hip_runtime.h>

// MergedEmbeddingBag (T tables, sum-pool, include_last_offset) + dense concat.
//
// Roofline: ~5.45 GB of random 512B embedding-row gathers (666 MB of tables
// >> 192 MB L2) at 23.3 TB/s -> ~240us floor; 122 MFLOP of adds is noise.
// No matrix op exists in this workload, so the CDNA5-specific path used is the
// async gather engine: GLOBAL_LOAD_ASYNC_TO_LDS_B128 (ASYNCcnt,
// CDNA5 ISA) queues row gathers into LDS with zero VGPR
// pressure. Offsets ride the scalar path; per-stage indices are fetched with a
// single coalesced vector load and broadcast via v_readlane_b32.

#define WAVES_PER_BLOCK 4
#define STAGE_ROWS 16   // rows staged into LDS per wait; L=10 fits in one stage

__device__ __forceinline__ void async_row16_to_lds(const void* gptr, unsigned lds_byte_addr) {
  // One instruction, 32 lanes x 16B = one 512B embedding row -> LDS.
  asm volatile("global_load_async_to_lds_b128 %0, %1, off"
               :: "v"(lds_byte_addr), "v"(gptr)
               : "memory");
}

__device__ __forceinline__ void wait_asynccnt0() {
#if __has_builtin(__builtin_amdgcn_s_wait_asynccnt)
  __builtin_amdgcn_s_wait_asynccnt(0);
#else
  asm volatile("s_wait_asynccnt 0" ::: "memory");
#endif
}

__device__ __forceinline__ int lane_bcast(int v, int lane) {
#if __has_builtin(__builtin_amdgcn_readlane)
  return __builtin_amdgcn_readlane(v, lane);   // v_readlane_b32 -> SGPR
#else
  return __shfl(v, lane, 32);
#endif
}

// grid: (ceil(B/WAVES_PER_BLOCK), T+1, 1); block: WAVES_PER_BLOCK*32
__global__ __launch_bounds__(WAVES_PER_BLOCK * 32) void
merged_embedding_bag_kernel(const int*   __restrict__ index,    // [T, N]
                            const int*   __restrict__ offsets,  // [T, B+1]
                            const float* __restrict__ dense,    // [B, 128]
                            const float* __restrict__ weights,  // [T, V, 128]
                            float*       __restrict__ out,      // [B, (T+1)*128]
                            int T, int N, int B, int V)
{
  __shared__ float4 stage[WAVES_PER_BLOCK][STAGE_ROWS][32]; // 32 KB

  const int lane   = threadIdx.x & 31;
  const int wave   = threadIdx.x >> 5;
  const int t_slot = blockIdx.y;                       // 0 = dense, 1..T = tables
  const int b      = blockIdx.x * WAVES_PER_BLOCK + wave;
  if (b >= B) return;
  // b is wave-uniform; pin it to the scalar unit so the offset loads go s_load.
  const int b_u = __builtin_amdgcn_readfirstlane(b);

  float4* outp =
      (float4*)(out + ((size_t)b_u * (size_t)(T + 1) + (size_t)t_slot) * 128) + lane;

  if (t_slot == 0) {  // uniform across the whole gridDim.y==0 slice: no divergence
    *outp = ((const float4*)(dense + (size_t)b_u * 128))[lane];
    return;
  }

  const int t = t_slot - 1;
  const int* __restrict__ off = offsets + (size_t)t * (size_t)(B + 1);
  const int r0 = __builtin_amdgcn_readfirstlane(off[b_u]);
  const int r1 = __builtin_amdgcn_readfirstlane(off[b_u + 1]);
  const int* __restrict__ idx = index + (size_t)t * (size_t)N;
  const float* __restrict__ Wt = weights + (size_t)t * (size_t)V * 128;

  // Workgroup-relative LDS byte address of this wave's stage buffer for this
  // lane: low 32 bits of the generic address of an LDS object are the LDS offset.
  const unsigned lds_base = (unsigned)(unsigned long long)(&stage[wave][0][lane]);

  float4 acc = make_float4(0.f, 0.f, 0.f, 0.f);

  for (int base = r0; base < r1; base += STAGE_ROWS) {
    const int nrows = min(STAGE_ROWS, r1 - base);

    // One coalesced vector load fetches this stage's indices (lanes 0..15 each
    // hold one, clamped to the bag end; upper lanes duplicate).
    const int pos  = min(base + (lane & (STAGE_ROWS - 1)), r1 - 1);
    const int ivec = idx[pos];

    // Issue all row gathers for this stage asynchronously (ASYNCcnt-tracked).
    for (int j = 0; j < nrows; ++j) {
      const int row = lane_bcast(ivec, j);             // index j -> scalar unit
      const float4* src = (const float4*)(Wt + (size_t)row * 128) + lane;
      const unsigned dst = lds_base + (unsigned)(j * 32 * (int)sizeof(float4));
      async_row16_to_lds((const void*)src, dst);
    }

    wait_asynccnt0();  // all staged rows landed in LDS

    for (int j = 0; j < nrows; ++j) {
      const float4 v = stage[wave][j][lane];
      acc.x += v.x; acc.y += v.y; acc.z += v.z; acc.w += v.w;
    }
  }

  *outp = acc;
}

extern "C" void kernel_launch(void* const* d_in, const int* in_sizes, int n_in,
                              void* d_out, int out_size, void* d_ws, size_t ws_size,
                              hipStream_t stream) {
  const int*   index   = (const int*)d_in[0];   // [T, N]
  const int*   offsets = (const int*)d_in[1];   // [T, B+1]
  const float* dense   = (const float*)d_in[2]; // [B, D]
  const float* weights = (const float*)d_in[3]; // [T, V, D]
  float*       out     = (float*)d_out;         // [B, (T+1)*D]

  const long S0 = in_sizes[0];
  const long S1 = in_sizes[1];
  const long S2 = in_sizes[2];
  const long S3 = in_sizes[3];

  const int Tp1 = (int)((long)out_size / S2);   // (T+1) since out = B*(T+1)*D, S2 = B*D
  const int T   = Tp1 - 1;
  const int N   = (int)(S0 / T);
  const int B   = (int)(S1 / T) - 1;
  const int D   = (int)(S2 / B);                // reference: D == 128
  const int V   = (int)(S3 / ((long)T * (long)D));
  (void)D; (void)n_in; (void)d_ws; (void)ws_size;

  dim3 grid((unsigned)((B + WAVES_PER_BLOCK - 1) / WAVES_PER_BLOCK), (unsigned)Tp1, 1);
  merged_embedding_bag_kernel<<<grid, WAVES_PER_BLOCK * 32, 0, stream>>>(
      index, offsets, dense, weights, out, T, N, B, V);
}